// SpatialDistanceGraph_63952063037633
// MI455X (gfx1250) — compile-verified
//
#include <hip/hip_runtime.h>
#include <hip/hip_bf16.h>

// ---------------- workspace layout (float offsets) ----------------
// dist[3][14][640]           : 26880 floats
// adj[14*14]                 : 196 floats
// EM  tiled (3 grids)        : 468224 floats  (exp(K), WMMA A-tile layout)
// EMT tiled (3 grids)        : 468224 floats  (exp(K)^T, same layout)
#define DIST_OFF 0
#define ADJ_OFF  26880
#define E_OFF    27136
#define ET_OFF   495360
#define ESZ      468224

typedef float v2f __attribute__((ext_vector_type(2)));
typedef float v8f __attribute__((ext_vector_type(8)));

__device__ __forceinline__ void pairIdx(int p, int& a, int& b) {
    int i = 0, rem = p, cnt = 13;           // D-1
    while (rem >= cnt) { rem -= cnt; ++i; --cnt; }
    a = i; b = i + 1 + rem;
}

// EM[i][k] = exp( exp(-10 * C(i,k)) ), zero-padded outside n x n
__device__ __forceinline__ float EMval(int g, int i, int k, int n) {
    if (i >= n || k >= n) return 0.0f;
    int xi = i % g, yi = i / g, xk = k % g, yk = k / g;
    float dx = (float)(xi - xk), dy = (float)(yi - yk);
    float c = sqrtf(dx * dx + dy * dy) * (1.0f / (1.41421356237f * (float)(g - 1)));
    float w = (yk < g / 3) ? 1.0f : ((yk < (2 * g) / 3) ? 1.0f : 1.2f);
    c *= w;
    float K = __expf(-10.0f * c);
    return __expf(K);
}

// ---------------- kernel 1: distributions + EM/EMT tiling + adj clear ----------------
__global__ void setup_kernel(const float* __restrict__ bb, float* __restrict__ ws) {
    const int G[3]   = {5, 15, 25};
    const int N[3]   = {25, 225, 625};
    const int NK4[3] = {8, 60, 160};
    const int EO[3]  = {0, 1024, 58624};
    const long long DN = 3LL * 14 * 640;                 // 26880
    const long long total = DN + 196 + 2LL * ESZ;
    for (long long idx = (long long)blockIdx.x * blockDim.x + threadIdx.x; idx < total;
         idx += (long long)gridDim.x * blockDim.x) {
        if (idx < DN) {
            int gi = (int)(idx / 8960);
            int rem = (int)(idx - (long long)gi * 8960);
            int d = rem / 640, i = rem - d * 640;
            int g = G[gi];
            float v = 0.0f;
            if (i < N[gi]) {
                float s = (float)g * 0.001f;
                float bx = bb[d * 4 + 0] * s, by = bb[d * 4 + 1] * s;
                float bw = bb[d * 4 + 2] * s, bh = bb[d * 4 + 3] * s;
                float x1 = fmaxf(truncf(bx), 0.0f), y1 = fmaxf(truncf(by), 0.0f);
                float x2 = fminf(truncf(bx + bw), (float)(g - 1));
                float y2 = fminf(truncf(by + bh), (float)(g - 1));
                int a = i / g, c = i % g;                 // a = grid row (y), c = col (x)
                bool nz = (bx != 0.0f) || (by != 0.0f) || (bw != 0.0f) || (bh != 0.0f);
                float fa = (float)a, fc = (float)c;
                v = (fa >= y1 && fa <= y2 && fc >= x1 && fc <= x2 && nz) ? 1.0f : 0.0f;
            }
            ws[DIST_OFF + (int)idx] = v;
        } else if (idx < DN + 196) {
            ws[ADJ_OFF + (int)(idx - DN)] = 0.0f;
        } else {
            long long e = idx - DN - 196;
            int isT = (e >= ESZ); if (isT) e -= ESZ;
            int gi = (e < 1024) ? 0 : ((e < 58624) ? 1 : 2);
            int local = (int)(e - EO[gi]);
            int blk = NK4[gi] * 64;
            int rt = local / blk;
            int rem = local - rt * blk;
            int ks = rem >> 6, q = rem & 63, L = q >> 1, c = q & 1;
            int i = rt * 16 + (L & 15);
            int k = ks * 4 + ((L >> 4) << 1) + c;
            float val = isT ? EMval(G[gi], k, i, N[gi]) : EMval(G[gi], i, k, N[gi]);
            ws[(isT ? ET_OFF : E_OFF) + EO[gi] + local] = val;
        }
    }
}

// ---------------- Sinkhorn half-update: dst = logSrc - log(M @ ew), ew = exp(dst) ----------------
__device__ __forceinline__ void sinkhorn_half(const float* __restrict__ Mt,
                                              const float* __restrict__ logSrc,
                                              float* __restrict__ dst,
                                              float* __restrict__ ew,
                                              int n, int NRT, int NK4, int lane, int wave) {
    float ust[3][8];
    int slot = 0;
    int pc = lane & 15, hb = lane >> 4;
    for (int rt = wave; rt < NRT; rt += 16, ++slot) {
        v8f acc = {0.f, 0.f, 0.f, 0.f, 0.f, 0.f, 0.f, 0.f};
        const float* Ew = Mt + (size_t)rt * NK4 * 64;
        if (rt + 16 < NRT) __builtin_prefetch(Mt + (size_t)(rt + 16) * NK4 * 64, 0, 0);
        for (int ks = 0; ks < NK4; ++ks) {
            v2f a = ((const v2f*)(Ew + ks * 64))[lane];
            int kb = (ks * 4 + (hb << 1)) * 16 + pc;
            v2f b;
            b.x = ew[kb];
            b.y = ew[kb + 16];
            acc = __builtin_amdgcn_wmma_f32_16x16x4_f32(false, a, false, b, (short)0, acc,
                                                        false, false);
        }
        int ibase = rt * 16 + (hb << 3);
#pragma unroll
        for (int r = 0; r < 8; ++r) {
            int i = ibase + r;
            ust[slot][r] = (i < n) ? (logSrc[i * 16 + pc] - __logf(acc[r] + 1e-38f))
                                   : -30000.0f;
        }
    }
    __syncthreads();   // everyone done reading ew
    slot = 0;
    for (int rt = wave; rt < NRT; rt += 16, ++slot) {
        int ibase = rt * 16 + (hb << 3);
#pragma unroll
        for (int r = 0; r < 8; ++r) {
            int i = ibase + r;
            float uu = ust[slot][r];
            dst[i * 16 + pc] = uu;
            ew[i * 16 + pc] = __expf(uu);
        }
    }
    __syncthreads();
}

// ---------------- kernel 2: persistent Sinkhorn, one WG per (pair-tile, grid) ----------------
__global__ __launch_bounds__(512) void sinkhorn_kernel(float* __restrict__ ws) {
    extern __shared__ float sm[];
    const int G[3]   = {5, 15, 25};
    const int N[3]   = {25, 225, 625};
    const int NTs[3] = {32, 240, 640};
    const int NK4s[3]= {8, 60, 160};
    const int EO[3]  = {0, 1024, 58624};
    const float SW[3]= {0.2f, 0.3f, 0.5f};

    int gi = blockIdx.y;
    int g = G[gi], n = N[gi], NT = NTs[gi], NK4 = NK4s[gi], NRT = NT >> 4;
    const float* EM  = ws + E_OFF + EO[gi];
    const float* EMT = ws + ET_OFF + EO[gi];

    float* logP = sm;              // 640*16
    float* logQ = sm + 10240;
    float* uA   = sm + 20480;
    float* vA   = sm + 30720;
    float* ew   = sm + 40960;
    float* emdb = sm + 51200;      // 16

    int tid = threadIdx.x, lane = tid & 31, wave = tid >> 5;
    int p0 = blockIdx.x * 16;

    // init logP/logQ/ew (padded pairs behave like uniform d=1 -> finite, discarded later)
    for (int idx = tid; idx < NT * 16; idx += 512) {
        int i = idx >> 4, pp = idx & 15, p = p0 + pp;
        float lp = 0.0f, lq = 0.0f;
        if (p < 91 && i < n) {
            int a, b; pairIdx(p, a, b);
            lp = __logf(ws[DIST_OFF + (gi * 14 + a) * 640 + i] + 1e-8f);
            lq = __logf(ws[DIST_OFF + (gi * 14 + b) * 640 + i] + 1e-8f);
        }
        logP[idx] = lp; logQ[idx] = lq; ew[idx] = 1.0f;   // exp(v=0)
    }
    __syncthreads();

    for (int it = 0; it < 100; ++it) {
        sinkhorn_half(EM,  logP, uA, ew, n, NRT, NK4, lane, wave);  // u update
        sinkhorn_half(EMT, logQ, vA, ew, n, NRT, NK4, lane, wave);  // v update
    }

    // EMD = sum_{i,j} exp(u_i + v_j + K_ij) * C_ij
    if (tid < 16) emdb[tid] = 0.0f;
    __syncthreads();
    float accp[16];
#pragma unroll
    for (int pp = 0; pp < 16; ++pp) accp[pp] = 0.0f;
    float invmax = 1.0f / (1.41421356237f * (float)(g - 1));
    int t1 = g / 3, t2 = (2 * g) / 3;
    for (int f = tid; f < n * n; f += 512) {
        int i = f / n, j = f - i * n;
        int xi = i % g, yi = i / g, xj = j % g, yj = j / g;
        float dx = (float)(xi - xj), dy = (float)(yi - yj);
        float C = sqrtf(dx * dx + dy * dy) * invmax;
        C *= (yj < t1) ? 1.0f : ((yj < t2) ? 1.0f : 1.2f);
        float K = __expf(-10.0f * C);
#pragma unroll
        for (int pp = 0; pp < 16; ++pp)
            accp[pp] += __expf(uA[i * 16 + pp] + vA[j * 16 + pp] + K) * C;
    }
#pragma unroll
    for (int pp = 0; pp < 16; ++pp) atomicAdd(&emdb[pp], accp[pp]);
    __syncthreads();
    if (tid < 16) {
        int p = p0 + tid;
        if (p < 91) {
            int a, b; pairIdx(p, a, b);
            float e = SW[gi] * emdb[tid];
            atomicAdd(&ws[ADJ_OFF + a * 14 + b], e);
            atomicAdd(&ws[ADJ_OFF + b * 14 + a], e);
        }
    }
}

// ---------------- kernel 3: adjacency softmax + embed + 3 GAT layers + output ----------------
__device__ __forceinline__ float blockReduceSum(float v, float* red) {
    int t = threadIdx.x;
    red[t] = v; __syncthreads();
#pragma unroll
    for (int s = 128; s > 0; s >>= 1) { if (t < s) red[t] += red[t + s]; __syncthreads(); }
    float r = red[0]; __syncthreads();
    return r;
}

__device__ __forceinline__ void layernorm256(float* buf, const float* g, const float* b,
                                             float* red, bool elu) {
    int t = threadIdx.x;
    for (int d = 0; d < 14; ++d) {
        float v = buf[d * 256 + t];
        float mean = blockReduceSum(v, red) * (1.0f / 256.0f);
        float dv = v - mean;
        float var = blockReduceSum(dv * dv, red) * (1.0f / 256.0f);
        float o = dv * rsqrtf(var + 1e-5f) * g[t] + b[t];
        if (elu) o = (o > 0.0f) ? o : (__expf(o) - 1.0f);
        buf[d * 256 + t] = o;
    }
    __syncthreads();
}

__global__ __launch_bounds__(256) void gat_kernel(
    const float* __restrict__ features, const float* __restrict__ w_embed,
    const float* __restrict__ b_embed, const float* __restrict__ g1,
    const float* __restrict__ b1, const float* __restrict__ gat_W,
    const float* __restrict__ gat_b, const float* __restrict__ gat_g,
    const float* __restrict__ gat_bt, const float* __restrict__ w_out,
    const float* __restrict__ b_out, const float* __restrict__ g2,
    const float* __restrict__ b2, const float* __restrict__ ws,
    float* __restrict__ out) {
    extern __shared__ float sm[];
    float* xs    = sm;            // 14*256
    float* hs    = sm + 3584;     // 14*256
    float* outs  = sm + 7168;     // 14*768
    float* adjs  = sm + 17920;    // 196
    float* attns = sm + 18116;    // 196
    float* fattn = sm + 18312;    // 196
    float* red   = sm + 18508;    // 256
    int t = threadIdx.x;

    // adjacency softmax
    if (t < 14) {
        const float* arow = ws + ADJ_OFF + t * 14;
        float mx = -1e30f;
        for (int j = 0; j < 14; ++j) mx = fmaxf(mx, arow[j]);
        float s = 0.0f; float tmp[14];
        for (int j = 0; j < 14; ++j) { tmp[j] = __expf(arow[j] - mx); s += tmp[j]; }
        float inv = 1.0f / s;
        for (int j = 0; j < 14; ++j) adjs[t * 14 + j] = tmp[j] * inv;
    }
    if (t < 196) fattn[t] = 0.0f;
    __syncthreads();

    // embed: x = LN(features @ w_embed + b_embed)
    {
        int c = t;
        for (int d = 0; d < 14; ++d) {
            float acc = b_embed[c];
            const float* fr = features + d * 768;
            for (int k = 0; k < 768; ++k) acc += fr[k] * w_embed[k * 256 + c];
            xs[d * 256 + c] = acc;
        }
    }
    __syncthreads();
    layernorm256(xs, g1, b1, red, false);

    for (int l = 0; l < 3; ++l) {
        const float* W = gat_W + l * 256 * 256;
        const float* bl = gat_b + l * 256;
        int c = t;
        for (int d = 0; d < 14; ++d) {
            float acc = bl[c];
            const float* xr = xs + d * 256;
            for (int k = 0; k < 256; ++k) acc += xr[k] * W[k * 256 + c];
            hs[d * 256 + c] = acc;
        }
        __syncthreads();
        if (t < 196) {
            int i = t / 14, j = t % 14;
            const float* hi = hs + i * 256;
            const float* hj = hs + j * 256;
            float s = 0.0f;
            for (int k = 0; k < 256; ++k) s += hi[k] * hj[k];
            s *= 0.0625f;                              // hd^-0.5, hd=256
            if (adjs[t] == 0.0f) s = -1e30f;
            attns[t] = s;
        }
        __syncthreads();
        if (t < 14) {
            float mx = -1e30f;
            for (int j = 0; j < 14; ++j) mx = fmaxf(mx, attns[t * 14 + j]);
            float s = 0.0f;
            for (int j = 0; j < 14; ++j) { float e = __expf(attns[t * 14 + j] - mx); attns[t * 14 + j] = e; s += e; }
            float inv = 1.0f / s;
            for (int j = 0; j < 14; ++j) attns[t * 14 + j] *= inv;
        }
        __syncthreads();
        if (t < 196) fattn[t] += attns[t] * (1.0f / 3.0f);
        float orow[14];
        for (int d = 0; d < 14; ++d) {
            float acc = 0.0f;
            for (int m = 0; m < 14; ++m) acc += attns[d * 14 + m] * hs[m * 256 + c];
            orow[d] = acc;
        }
        __syncthreads();
        for (int d = 0; d < 14; ++d) xs[d * 256 + c] = orow[d];
        __syncthreads();
        layernorm256(xs, gat_g + l * 256, gat_bt + l * 256, red, true);  // LN then elu
    }

    // out = LN(x @ w_out + b_out)
    for (int cc = t; cc < 768; cc += 256) {
        for (int d = 0; d < 14; ++d) {
            float acc = b_out[cc];
            const float* xr = xs + d * 256;
            for (int k = 0; k < 256; ++k) acc += xr[k] * w_out[k * 768 + cc];
            outs[d * 768 + cc] = acc;
        }
    }
    __syncthreads();
    for (int d = 0; d < 14; ++d) {
        float part = 0.0f;
        for (int cc = t; cc < 768; cc += 256) part += outs[d * 768 + cc];
        float mean = blockReduceSum(part, red) * (1.0f / 768.0f);
        float vp = 0.0f;
        for (int cc = t; cc < 768; cc += 256) { float dv = outs[d * 768 + cc] - mean; vp += dv * dv; }
        float var = blockReduceSum(vp, red) * (1.0f / 768.0f);
        float inv = rsqrtf(var + 1e-5f);
        for (int cc = t; cc < 768; cc += 256)
            out[d * 768 + cc] = (outs[d * 768 + cc] - mean) * inv * g2[cc] + b2[cc];
    }
    __syncthreads();
    if (t < 196) out[10752 + t] = fattn[t];
}

extern "C" void kernel_launch(void* const* d_in, const int* in_sizes, int n_in,
                              void* d_out, int out_size, void* d_ws, size_t ws_size,
                              hipStream_t stream) {
    const float* features = (const float*)d_in[0];
    const float* bb       = (const float*)d_in[1];
    const float* w_embed  = (const float*)d_in[2];
    const float* b_embed  = (const float*)d_in[3];
    const float* g1       = (const float*)d_in[4];
    const float* b1       = (const float*)d_in[5];
    const float* gat_W    = (const float*)d_in[6];
    const float* gat_b    = (const float*)d_in[7];
    const float* gat_g    = (const float*)d_in[8];
    const float* gat_bt   = (const float*)d_in[9];
    const float* w_out    = (const float*)d_in[10];
    const float* b_out    = (const float*)d_in[11];
    const float* g2       = (const float*)d_in[12];
    const float* b2       = (const float*)d_in[13];
    float* ws = (float*)d_ws;
    float* out = (float*)d_out;

    setup_kernel<<<1024, 256, 0, stream>>>(bb, ws);
    // 6 pair-tiles x 3 grids, 512 threads (16 waves), 200 KB LDS for Sinkhorn state
    sinkhorn_kernel<<<dim3(6, 3), 512, 51216 * sizeof(float), stream>>>(ws);
    gat_kernel<<<1, 256, 18764 * sizeof(float), stream>>>(
        features, w_embed, b_embed, g1, b1, gat_W, gat_b, gat_g, gat_bt,
        w_out, b_out, g2, b2, ws, out);
}